// V42Loss_82188494176648
// MI455X (gfx1250) — compile-verified
//
#include <hip/hip_runtime.h>
#include <hip/hip_bf16.h>
#include <stdint.h>

typedef __attribute__((ext_vector_type(2))) float v2f;
typedef __attribute__((ext_vector_type(8))) float v8f;

#define NBLOCKS  1024
#define NTHREADS 256
#define TILE_B   64
#define BATCH    524288
#define NTILES   (BATCH / TILE_B)
#define PAR_STRIDE 32           // padded per-block partials row (24 used + 8 zero)

// LDS layout in floats (dynamic shared memory), double-buffered tiles
#define P_STRIDE 68             // 64 floats/row + pad (bank spread), 16B-aligned
#define T_STRIDE 36             // 8 floats x 4 steps + pad
#define Q_STRIDE 20             // 16 floats + pad
#define P_BUF    (TILE_B * P_STRIDE)              // 4352
#define T_BUF    (TILE_B * T_STRIDE)              // 2304
#define Q_BUF    (TILE_B * Q_STRIDE)              // 1280
#define P_OFF    0
#define T_OFF    (2 * P_BUF)                      // 8704
#define Q_OFF    (T_OFF + 2 * T_BUF)              // 13312
#define RED_OFF  (Q_OFF + 2 * Q_BUF)              // 15872
#define SMEM_FLOATS (RED_OFF + NTHREADS * 6)      // 17408 floats = 69632 B

// ---- CDNA5 async global->LDS helpers ---------------------------------------

__device__ __forceinline__ uint32_t lds_off(const void* p) {
  // Generic LDS address: addr[31:0] is the LDS byte offset (aperture in high).
  return (uint32_t)(uintptr_t)p;
}

__device__ __forceinline__ void async_b128(uint32_t lds_byte, const void* gbase,
                                           uint32_t gbyte_off) {
  // GVS form: mem_addr = SADDR(64) + VADDR(32); dest LDS byte addr in vdst.
  // Streamed-once data: non-temporal hint.
  asm volatile("global_load_async_to_lds_b128 %0, %1, %2 th:TH_LOAD_NT"
               :
               : "v"(lds_byte), "v"(gbyte_off), "s"((uint64_t)gbase)
               : "memory");
}

__device__ __forceinline__ void wait_async_le(int prev_in_flight_ok) {
  if (prev_in_flight_ok) {
    // 7 async ops per wave per tile; in-order completion => <=7 means the
    // previous tile's 7 ops are done while the next tile's 7 stay in flight.
    asm volatile("s_wait_asynccnt 7" ::: "memory");
  } else {
    asm volatile("s_wait_asynccnt 0" ::: "memory");
  }
}

// ---- Stage 1: streaming partial sums ---------------------------------------
// Per step s (0..3) accumulate 6 scalars over the batch:
//   [0] mse   = sum_b sum_v var_w[v] * (pred-target)^2 at dims {2,1,0,3,4}
//   [1] hnum  = sum_b (pd3 - pd4)^2        [2] hden = sum_b pd3^2
//   [3] frost = sum_b relu((q2-p2)*sc2)^2  [4] press = sum_b (.5*relu((q0-p0)*sc0))^2
//   [5] temp  = sum_b sum_{d in 5,7,8,9,11} relu(|p_d-q_d|*sc_d - 5)^2
__global__ void __launch_bounds__(NTHREADS)
loss_partials(const float* __restrict__ pred, const float* __restrict__ targ,
              const float* __restrict__ seq, const float* __restrict__ scale,
              const float* __restrict__ mean, float* __restrict__ partials) {
  extern __shared__ float smem[];

  const int t  = threadIdx.x;
  const int s  = t >> 6;   // step 0..3, uniform per wave pair
  const int bl = t & 63;   // batch-in-tile

  const float sc0 = scale[0], sc2 = scale[2], sc3 = scale[3], sc4 = scale[4];
  const float sc5 = scale[5], sc7 = scale[7], sc8 = scale[8], sc9 = scale[9];
  const float sc11 = scale[11];
  const float mn3 = mean[3], mn4 = mean[4];

  float a_mse = 0.f, a_hn = 0.f, a_hd = 0.f, a_fr = 0.f, a_pr = 0.f, a_tp = 0.f;

  // Issue one tile's staging (7 async b128 per thread) into buffer `buf`.
  auto issue_tile = [&](int tile, int buf) {
    const uint32_t base_b = (uint32_t)tile * TILE_B;
    float* pb = smem + P_OFF + buf * P_BUF;
    float* tb = smem + T_OFF + buf * T_BUF;
    float* qb = smem + Q_OFF + buf * Q_BUF;
    // pred: full rows, 1024 x b128 -> 4 per thread
#pragma unroll
    for (int k = 0; k < 4; ++k) {
      uint32_t c = (uint32_t)t + NTHREADS * k;
      uint32_t b = c >> 4, ch = c & 15;
      async_b128(lds_off(pb + b * P_STRIDE + ch * 4u), pred,
                 ((base_b + b) * 64u + ch * 4u) * 4u);
    }
    // target: dims 0..7 of each (b,s) row, 512 x b128 -> 2 per thread
#pragma unroll
    for (int k = 0; k < 2; ++k) {
      uint32_t c = (uint32_t)t + NTHREADS * k;
      uint32_t b = c >> 3, w = c & 7;
      uint32_t st = w >> 1, hf = w & 1;
      async_b128(lds_off(tb + b * T_STRIDE + st * 8u + hf * 4u), targ,
                 ((base_b + b) * 64u + st * 16u + hf * 4u) * 4u);
    }
    // input_sequence last row: 256 x b128 -> 1 per thread
    {
      uint32_t b = (uint32_t)t >> 2, ch = (uint32_t)t & 3;
      async_b128(lds_off(qb + b * Q_STRIDE + ch * 4u), seq,
                 ((base_b + b) * 128u + 112u + ch * 4u) * 4u);
    }
  };

  int buf = 0;
  issue_tile(blockIdx.x, 0);  // prologue

  for (int tile = blockIdx.x; tile < NTILES; tile += gridDim.x) {
    const int next = tile + gridDim.x;
    const bool has_next = next < NTILES;        // uniform per block
    if (has_next) issue_tile(next, buf ^ 1);
    wait_async_le(has_next ? 1 : 0);
    __syncthreads();                            // current buffer fully staged

    const float* pb = smem + P_OFF + buf * P_BUF;
    const float* tb = smem + T_OFF + buf * T_BUF;
    const float* qb = smem + Q_OFF + buf * Q_BUF;
    const float* P = pb + bl * P_STRIDE + s * 16;
    const float* T = tb + bl * T_STRIDE + s * 8;
    const float* Q = (s == 0) ? (qb + bl * Q_STRIDE)
                              : (pb + bl * P_STRIDE + (s - 1) * 16);

    float p0 = P[0], p1 = P[1], p2 = P[2], p3 = P[3], p4 = P[4];
    float d0 = p0 - T[0], d1 = p1 - T[1], d2 = p2 - T[2], d3 = p3 - T[3],
          d4 = p4 - T[4];
    a_mse += 0.5f * d2 * d2 + 0.15f * d1 * d1 + 0.15f * d0 * d0 +
             0.1f * d3 * d3 + 0.1f * d4 * d4;

    float pd3 = fmaf(p3, sc3, mn3), pd4 = fmaf(p4, sc4, mn4);
    float hb = pd3 - pd4;
    a_hn += hb * hb;
    a_hd += pd3 * pd3;

    float fr = fmaxf((Q[2] - p2) * sc2, 0.f);
    a_fr += fr * fr;
    float pr = 0.5f * fmaxf((Q[0] - p0) * sc0, 0.f);
    a_pr += pr * pr;

    float t5  = fmaxf(fabsf((P[5]  - Q[5])  * sc5)  - 5.f, 0.f);
    float t7  = fmaxf(fabsf((P[7]  - Q[7])  * sc7)  - 5.f, 0.f);
    float t8  = fmaxf(fabsf((P[8]  - Q[8])  * sc8)  - 5.f, 0.f);
    float t9  = fmaxf(fabsf((P[9]  - Q[9])  * sc9)  - 5.f, 0.f);
    float t11 = fmaxf(fabsf((P[11] - Q[11]) * sc11) - 5.f, 0.f);
    a_tp += t5 * t5 + t7 * t7 + t8 * t8 + t9 * t9 + t11 * t11;

    __syncthreads();  // all waves done reading buf before it is re-issued
    buf ^= 1;
  }

  // Block reduction within each 64-thread s-group
  float* red = smem + RED_OFF;
  red[t * 6 + 0] = a_mse;
  red[t * 6 + 1] = a_hn;
  red[t * 6 + 2] = a_hd;
  red[t * 6 + 3] = a_fr;
  red[t * 6 + 4] = a_pr;
  red[t * 6 + 5] = a_tp;
  __syncthreads();
  for (int off = 32; off >= 1; off >>= 1) {
    if (bl < off) {
#pragma unroll
      for (int k = 0; k < 6; ++k) red[t * 6 + k] += red[(t + off) * 6 + k];
    }
    __syncthreads();
  }
  if (bl == 0) {
#pragma unroll
    for (int k = 0; k < 6; ++k)
      partials[blockIdx.x * PAR_STRIDE + s * 6 + k] = red[t * 6 + k];
  }
  if (t < PAR_STRIDE - 24)  // zero the pad so finalize loads are branch-free
    partials[blockIdx.x * PAR_STRIDE + 24 + t] = 0.f;
}

// ---- Stage 2: WMMA reduction of NBLOCKSx32 partials + final scalar ---------
// D(16x16) += ones(16x4) x P_chunk(4x16): after NBLOCKS/4 chained WMMAs, row 0
// of D holds the column sums. Two column passes cover all 32 (24 live) columns.
// Single wave32, EXEC all ones, branch-free hot loop (pad made guards moot).
__global__ void __launch_bounds__(32)
finalize_loss(const float* __restrict__ partials, float* __restrict__ out) {
  const int lane = threadIdx.x;
  __shared__ float tot[24];

#if __has_builtin(__builtin_amdgcn_wmma_f32_16x16x4_f32)
  v2f a;  a[0] = 1.f;  a[1] = 1.f;            // A = ones (16x4)
  v8f dA = {0.f, 0.f, 0.f, 0.f, 0.f, 0.f, 0.f, 0.f};
  v8f dB = {0.f, 0.f, 0.f, 0.f, 0.f, 0.f, 0.f, 0.f};
  const int hi = lane >> 4;                   // K group
  const int n  = lane & 15;                   // column
  for (int it = 0; it < NBLOCKS / 4; ++it) {
    v2f b0, b1;
#pragma unroll
    for (int r = 0; r < 2; ++r) {
      int row = it * 4 + 2 * hi + r;          // K = 2*(lane/16) + vgpr
      b0[r] = partials[row * PAR_STRIDE + n];         // columns 0..15
      b1[r] = partials[row * PAR_STRIDE + 16 + n];    // columns 16..31 (pad 0)
    }
    dA = __builtin_amdgcn_wmma_f32_16x16x4_f32(false, a, false, b0, (short)0,
                                               dA, false, false);
    dB = __builtin_amdgcn_wmma_f32_16x16x4_f32(false, a, false, b1, (short)0,
                                               dB, false, false);
  }
  if (lane < 16) tot[lane] = dA[0];           // D[M=0][N=lane]
  if (lane < 8)  tot[16 + lane] = dB[0];
#else
  if (lane < 24) {
    float ssum = 0.f;
    for (int i = 0; i < NBLOCKS; ++i) ssum += partials[i * PAR_STRIDE + lane];
    tot[lane] = ssum;
  }
#endif
  __syncthreads();

  if (lane == 0) {
    const float invB = 1.f / (float)BATCH;
    const float H = 1.f + 0.5f + (1.f / 3.f) + 0.25f;  // sum 1/(s+1)
    float loss = 0.f;
#pragma unroll
    for (int st = 0; st < 4; ++st) {
      float w    = (1.f / (float)(st + 1)) / H;        // step_w[st]
      float mse  = tot[st * 6 + 0] * invB;
      float heat = (tot[st * 6 + 1] * invB) / (tot[st * 6 + 2] * invB + 1e-6f);
      float phys = heat + tot[st * 6 + 3] * invB + tot[st * 6 + 4] * invB +
                   0.1f * tot[st * 6 + 5] * invB;
      loss += w * (mse + 0.1f * phys);
    }
    out[0] = loss;
  }
}

// ---- Host launch -----------------------------------------------------------
extern "C" void kernel_launch(void* const* d_in, const int* in_sizes, int n_in,
                              void* d_out, int out_size, void* d_ws,
                              size_t ws_size, hipStream_t stream) {
  const float* pred  = (const float*)d_in[0];
  const float* targ  = (const float*)d_in[1];
  const float* seq   = (const float*)d_in[2];
  const float* scale = (const float*)d_in[3];
  const float* mean  = (const float*)d_in[4];
  float* out         = (float*)d_out;
  float* partials    = (float*)d_ws;  // NBLOCKS*PAR_STRIDE floats = 128 KB

  (void)in_sizes; (void)n_in; (void)out_size; (void)ws_size;

  loss_partials<<<NBLOCKS, NTHREADS, SMEM_FLOATS * sizeof(float), stream>>>(
      pred, targ, seq, scale, mean, partials);
  finalize_loss<<<1, 32, 0, stream>>>(partials, out);
}